// GAT_Layer_18605798326375
// MI455X (gfx1250) — compile-verified
//
#include <hip/hip_runtime.h>
#include <hip/hip_bf16.h>

typedef __attribute__((ext_vector_type(2))) float v2f;
typedef __attribute__((ext_vector_type(8))) float v8f;

#define KG_E_DIM 200000
#define IN_DIM 256
#define OUT_DIM 128

// ---------------------------------------------------------------------------
// Kernel 1: gat_in[KG_E,128] = ent_embed[KG_E,256] @ W[128,256]^T + b
// fp32 WMMA 16x16x4. Block = 256 threads (8 waves). blockIdx.y selects which
// 64-column half of W (staged in 64KB LDS). Each wave computes a 16x64 strip.
// ---------------------------------------------------------------------------
__global__ void __launch_bounds__(256) gat_gemm_kernel(
    const float* __restrict__ A,      // [KG_E, 256]
    const float* __restrict__ W,      // [128, 256]
    const float* __restrict__ bias,   // [128]
    float* __restrict__ out,          // [KG_E, 128]
    int n_rows) {
  __shared__ float wlds[64 * IN_DIM];  // 64 output cols x 256 k  (64 KB)

  const int tid = threadIdx.x;
  const int colBase = blockIdx.y * 64;

  // Stage this half of W into LDS (contiguous 16384 floats).
  {
    const float4* src = (const float4*)(W + colBase * IN_DIM);
    float4* dst = (float4*)wlds;
    #pragma unroll
    for (int i = 0; i < (64 * IN_DIM / 4) / 256; ++i)
      dst[tid + i * 256] = src[tid + i * 256];
  }
  __syncthreads();

  const int wave = tid >> 5;
  const int lane = tid & 31;
  const int m = lane & 15;              // M (for A) / N (for B) within tile
  const int kphase = (lane < 16) ? 0 : 2;  // lanes 16-31 hold K+2,K+3

  const int row0 = blockIdx.x * 128 + wave * 16;
  if (row0 >= n_rows) return;           // KG_E divisible by 16 -> full tiles

  v8f acc[4] = {};
  const float* arow = A + (row0 + m) * IN_DIM + kphase;

  #pragma unroll 4
  for (int k = 0; k < IN_DIM; k += 4) {
    v2f a = *(const v2f*)(arow + k);
    #pragma unroll
    for (int t = 0; t < 4; ++t) {
      v2f b = *(const v2f*)(&wlds[(t * 16 + m) * IN_DIM + k + kphase]);
      acc[t] = __builtin_amdgcn_wmma_f32_16x16x4_f32(
          false, a, false, b, (short)0, acc[t], false, false);
    }
  }

  // C/D layout: lanes 0-15 -> M = r, lanes 16-31 -> M = r + 8; N = lane & 15.
  const int mrow = (lane < 16) ? 0 : 8;
  #pragma unroll
  for (int t = 0; t < 4; ++t) {
    const int col = colBase + t * 16 + m;
    const float bv = bias[col];
    #pragma unroll
    for (int r = 0; r < 8; ++r) {
      out[(row0 + mrow + r) * OUT_DIM + col] = acc[t][r] + bv;
    }
  }
}

// ---------------------------------------------------------------------------
// Kernel 2: per-node attention half-scores.
//   rscore[i] = gat_in[i]        . w_atten[128:256]   for i in [0, KG_E)
//   lscore[j] = gat_in[bids[j]]  . w_atten[0:128]     for j in [0, B)
// One wave per node: lane handles 4 floats, shuffle-reduce.
// ---------------------------------------------------------------------------
__global__ void __launch_bounds__(256) score_kernel(
    const float* __restrict__ gat_in,
    const int* __restrict__ batch_ids,
    const float* __restrict__ w_atten,   // [256]
    float* __restrict__ rscore,          // [KG_E]
    float* __restrict__ lscore,          // [B]
    int n_nodes, int n_batch) {
  const int wid = blockIdx.x * 8 + (threadIdx.x >> 5);
  const int lane = threadIdx.x & 31;

  float d = 0.0f;
  if (wid < n_nodes) {
    const float4 g = *(const float4*)(gat_in + (size_t)wid * OUT_DIM + lane * 4);
    const float4 w = *(const float4*)(w_atten + OUT_DIM + lane * 4);
    d = g.x * w.x + g.y * w.y + g.z * w.z + g.w * w.w;
    #pragma unroll
    for (int off = 16; off > 0; off >>= 1) d += __shfl_down(d, off, 32);
    if (lane == 0) rscore[wid] = d;
  } else if (wid < n_nodes + n_batch) {
    const int j = wid - n_nodes;
    const int idx = batch_ids[j];
    const float4 g = *(const float4*)(gat_in + (size_t)idx * OUT_DIM + lane * 4);
    const float4 w = *(const float4*)(w_atten + lane * 4);
    d = g.x * w.x + g.y * w.y + g.z * w.z + g.w * w.w;
    #pragma unroll
    for (int off = 16; off > 0; off >>= 1) d += __shfl_down(d, off, 32);
    if (lane == 0) lscore[j] = d;
  }
}

// ---------------------------------------------------------------------------
// Kernel 3: per-edge attention + scatter-accumulate.
// One wave per edge; lane covers 4 of the 128 dims. gat_in gathers hit L2
// (102 MB < 192 MB); accumulators (10 MB) are L2-resident f32 atomics.
// ---------------------------------------------------------------------------
__global__ void __launch_bounds__(256) edge_kernel(
    const float* __restrict__ gat_in,
    const int* __restrict__ rows,
    const int* __restrict__ cols,
    const float* __restrict__ lscore,
    const float* __restrict__ rscore,
    float* __restrict__ accum,           // [B, 128]
    float* __restrict__ rowsum,          // [B]
    int n_edges) {
  const int e = blockIdx.x * 8 + (threadIdx.x >> 5);
  if (e >= n_edges) return;
  const int lane = threadIdx.x & 31;

  const int row = rows[e];
  const int col = cols[e];
  const float s = lscore[row] + rscore[col];
  const float lr = (s >= 0.0f) ? s : 0.2f * s;   // leaky_relu(0.2)
  const float att = __expf(-lr);

  const float4 g = *(const float4*)(gat_in + (size_t)col * OUT_DIM + lane * 4);
  float* dst = accum + (size_t)row * OUT_DIM + lane * 4;
  unsafeAtomicAdd(dst + 0, att * g.x);
  unsafeAtomicAdd(dst + 1, att * g.y);
  unsafeAtomicAdd(dst + 2, att * g.z);
  unsafeAtomicAdd(dst + 3, att * g.w);
  if (lane == 0) unsafeAtomicAdd(rowsum + row, att);
}

// ---------------------------------------------------------------------------
// Kernel 4: out = prelu(accum / rowsum)
// ---------------------------------------------------------------------------
__global__ void __launch_bounds__(256) finalize_kernel(
    const float* __restrict__ accum,
    const float* __restrict__ rowsum,
    const float* __restrict__ prelu_w,
    float* __restrict__ out,
    int n) {
  const int i = blockIdx.x * 256 + threadIdx.x;
  if (i >= n) return;
  const float x = accum[i] / rowsum[i >> 7];
  out[i] = (x >= 0.0f) ? x : prelu_w[0] * x;
}

// ---------------------------------------------------------------------------
// Kernel 0: zero the atomic accumulators (deterministic per call).
// ---------------------------------------------------------------------------
__global__ void __launch_bounds__(256) zero_kernel(
    float* __restrict__ accum, float* __restrict__ rowsum, int n_acc, int n_rs) {
  const int i = blockIdx.x * 256 + threadIdx.x;
  if (i < n_acc) accum[i] = 0.0f;
  if (i < n_rs) rowsum[i] = 0.0f;
}

extern "C" void kernel_launch(void* const* d_in, const int* in_sizes, int n_in,
                              void* d_out, int out_size, void* d_ws, size_t ws_size,
                              hipStream_t stream) {
  const float* ent_embed = (const float*)d_in[0];  // [KG_E, 256]
  const int*   batch_ids = (const int*)d_in[1];    // [B]
  const int*   edge_idx  = (const int*)d_in[2];    // [2, E]
  const float* W         = (const float*)d_in[3];  // [128, 256]
  const float* b         = (const float*)d_in[4];  // [128]
  const float* w_atten   = (const float*)d_in[5];  // [256]
  const float* prelu_w   = (const float*)d_in[6];  // [1]

  const int kg_e = in_sizes[0] / IN_DIM;   // 200000
  const int nB   = in_sizes[1];            // 20000
  const int E    = in_sizes[2] / 2;        // 640000
  const int* rows = edge_idx;
  const int* cols = edge_idx + E;

  // Workspace layout (floats)
  float* ws      = (float*)d_ws;
  float* gat_in  = ws;                                   // KG_E * 128
  float* rscore  = gat_in + (size_t)kg_e * OUT_DIM;      // KG_E
  float* lscore  = rscore + kg_e;                        // B
  float* rowsum  = lscore + nB;                          // B
  float* accum   = rowsum + nB;                          // B * 128

  const int n_acc = nB * OUT_DIM;

  // 0) zero accumulators
  {
    int n = (n_acc > nB) ? n_acc : nB;
    zero_kernel<<<(n + 255) / 256, 256, 0, stream>>>(accum, rowsum, n_acc, nB);
  }
  // 1) GEMM: gat_in = ent_embed @ W^T + b  (fp32 WMMA)
  {
    dim3 grid((kg_e + 127) / 128, 2);
    gat_gemm_kernel<<<grid, 256, 0, stream>>>(ent_embed, W, b, gat_in, kg_e);
  }
  // 2) per-node half-scores
  {
    int nwaves = kg_e + nB;
    score_kernel<<<(nwaves + 7) / 8, 256, 0, stream>>>(
        gat_in, batch_ids, w_atten, rscore, lscore, kg_e, nB);
  }
  // 3) per-edge attention + scatter
  {
    edge_kernel<<<(E + 7) / 8, 256, 0, stream>>>(
        gat_in, rows, cols, lscore, rscore, accum, rowsum, E);
  }
  // 4) normalize + PReLU -> d_out
  {
    finalize_kernel<<<(n_acc + 255) / 256, 256, 0, stream>>>(
        accum, rowsum, prelu_w, (float*)d_out, n_acc);
  }
}